// HungarianMatcher_49581102465467
// MI455X (gfx1250) — compile-verified
//
#include <hip/hip_runtime.h>
#include <hip/hip_bf16.h>
#include <math.h>

#define TWO_PI_F 6.283185307179586f
#define PI_F     3.141592653589793f

// Problem dims (fixed by reference setup_inputs)
#define BB 2
#define CC 10
#define NN 100
#define MM 50
#define HP 24
#define WP 24
#define HH 96
#define WW 96
#define KK (HH * WW)   // 9216

typedef __attribute__((ext_vector_type(2))) float v2f;
typedef __attribute__((ext_vector_type(8))) float v8f;

// ---------------------------------------------------------------------------
// Kernel 1: per (b,n) — bilinear resize 24x24 -> 96x96 (JAX half-pixel,
// edge-clamped), emit P row and D = f1(p)-f0(p) row, reduce sumP and F0.
// ---------------------------------------------------------------------------
__global__ __launch_bounds__(256) void prep_pred_kernel(
    const float* __restrict__ pred_mask,   // B*N*24*24
    float* __restrict__ P,                 // B*N*9216
    float* __restrict__ Dm,                // B*N*9216
    float* __restrict__ sumP,              // B*N
    float* __restrict__ F0)                // B*N
{
    const int bn = blockIdx.x;             // 0 .. B*N-1
    __shared__ float s[HP * WP];
    __shared__ float redS[256];
    __shared__ float redF[256];

    const float* src = pred_mask + (size_t)bn * (HP * WP);
    for (int i = threadIdx.x; i < HP * WP; i += 256) s[i] = src[i];
    __syncthreads();

    float ls = 0.0f, lf = 0.0f;
    float* Pr = P  + (size_t)bn * KK;
    float* Dr = Dm + (size_t)bn * KK;

    for (int pix = threadIdx.x; pix < KK; pix += 256) {
        const int oy = pix / WW;
        const int ox = pix - oy * WW;
        // input coord = (out + 0.5)/scale - 0.5, scale = 4
        const float fy = (oy + 0.5f) * 0.25f - 0.5f;
        const float fx = (ox + 0.5f) * 0.25f - 0.5f;
        int y0 = (int)floorf(fy);
        int x0 = (int)floorf(fx);
        const float wy = fy - (float)y0;
        const float wx = fx - (float)x0;
        const int y0c = max(y0, 0),     x0c = max(x0, 0);
        const int y1c = min(y0 + 1, HP - 1), x1c = min(x0 + 1, WP - 1);
        const float v00 = s[y0c * WP + x0c];
        const float v01 = s[y0c * WP + x1c];
        const float v10 = s[y1c * WP + x0c];
        const float v11 = s[y1c * WP + x1c];
        const float p = (1.0f - wy) * ((1.0f - wx) * v00 + wx * v01)
                      +          wy * ((1.0f - wx) * v10 + wx * v11);

        const float pc  = fminf(fmaxf(p, 0.001f), 0.999f);
        const float omc = 1.0f - pc;
        // t==1 focal contribution: 0.25 * (-log pc) * (1-pc)^2
        const float f1 = -0.25f * logf(pc) * omc * omc;
        // t==0 focal contribution: 0.75 * (-log(1-pc)) * pc^2
        const float f0 = -0.75f * logf(omc) * pc * pc;

        Pr[pix] = p;
        Dr[pix] = f1 - f0;
        ls += fabsf(p);
        lf += f0;
    }

    redS[threadIdx.x] = ls;
    redF[threadIdx.x] = lf;
    __syncthreads();
    for (int sft = 128; sft > 0; sft >>= 1) {
        if ((int)threadIdx.x < sft) {
            redS[threadIdx.x] += redS[threadIdx.x + sft];
            redF[threadIdx.x] += redF[threadIdx.x + sft];
        }
        __syncthreads();
    }
    if (threadIdx.x == 0) {
        sumP[bn] = redS[0];
        F0[bn]   = redF[0];
    }
}

// ---------------------------------------------------------------------------
// Kernel 2: per (b,m) — count1 = sum |target_mask|
// ---------------------------------------------------------------------------
__global__ __launch_bounds__(256) void prep_tgt_kernel(
    const float* __restrict__ tmask,       // B*M*9216
    float* __restrict__ count1)            // B*M
{
    const int bm = blockIdx.x;
    __shared__ float red[256];
    const float* src = tmask + (size_t)bm * KK;
    float ls = 0.0f;
    for (int i = threadIdx.x; i < KK; i += 256) ls += fabsf(src[i]);
    red[threadIdx.x] = ls;
    __syncthreads();
    for (int sft = 128; sft > 0; sft >>= 1) {
        if ((int)threadIdx.x < sft) red[threadIdx.x] += red[threadIdx.x + sft];
        __syncthreads();
    }
    if (threadIdx.x == 0) count1[bm] = red[0];
}

// ---------------------------------------------------------------------------
// Kernel 3: one wave32 per 16x16 output tile.
//   Two fp32 WMMA GEMM chains (S1 = P·Tᵗ, G = D·Tᵗ) sharing the B matrix,
//   then fused epilogue: dice / focal / class / momentum-l1 / eIoU.
// A (16x4 f32): lane&15 = row, float2 at k + 2*(lane>>4).
// B (4x16 f32): lane&15 = col, float2 at k + 2*(lane>>4) from row-major T.
// D (16x16 f32): acc[r] -> row r + 8*(lane>>4), col lane&15.
// ---------------------------------------------------------------------------
__global__ __launch_bounds__(32) void cost_kernel(
    const float* __restrict__ pred_prob,       // B*C*N
    const float* __restrict__ pred_momentum,   // B*N*4
    const int*   __restrict__ target_class,    // B*M
    const float* __restrict__ target_mask,     // B*M*9216 (binary, used as B matrix)
    const float* __restrict__ target_momentum, // B*M*4
    const float* __restrict__ P,
    const float* __restrict__ Dm,
    const float* __restrict__ sumP,
    const float* __restrict__ F0,
    const float* __restrict__ count1,
    float* __restrict__ out)                   // B*N*M
{
    const int tm = blockIdx.x;       // 0..3  (M tiles)
    const int tn = blockIdx.y;       // 0..6  (N tiles)
    const int b  = blockIdx.z;       // 0..1
    const int lane = threadIdx.x;    // 0..31
    const int kh = lane >> 4;        // which K-half this lane loads
    const int lr = lane & 15;

    // A rows (clamped: pad rows only affect pad outputs, never written)
    const int an  = tn * 16 + lr;
    const int anc = min(an, NN - 1);
    const float* __restrict__ Prow = P  + ((size_t)b * NN + anc) * KK;
    const float* __restrict__ Drow = Dm + ((size_t)b * NN + anc) * KK;

    // B cols (clamped likewise)
    const int bm  = tm * 16 + lr;
    const int bmc = min(bm, MM - 1);
    const float* __restrict__ Trow = target_mask + ((size_t)b * MM + bmc) * KK;

    const int ko = kh * 2;

    v8f accS = {0.f, 0.f, 0.f, 0.f, 0.f, 0.f, 0.f, 0.f};
    v8f accG = {0.f, 0.f, 0.f, 0.f, 0.f, 0.f, 0.f, 0.f};

    #pragma unroll 4
    for (int k = 0; k < KK; k += 4) {
        const v2f a  = *(const v2f*)(Prow + k + ko);
        const v2f d  = *(const v2f*)(Drow + k + ko);
        const v2f bt = *(const v2f*)(Trow + k + ko);
        accS = __builtin_amdgcn_wmma_f32_16x16x4_f32(
            false, a, false, bt, (short)0, accS, false, false);
        accG = __builtin_amdgcn_wmma_f32_16x16x4_f32(
            false, d, false, bt, (short)0, accG, false, false);
    }

    // ---------------- epilogue (divergence fine after WMMA loop) -----------
    const int m = tm * 16 + lr;
    if (m >= MM) return;

    const int   cls = target_class[b * MM + m];
    const float c1  = count1[b * MM + m];
    const float* tmo = target_momentum + ((size_t)b * MM + m) * 4;
    const float t0 = tmo[0], t1 = tmo[1], t2 = tmo[2], t3 = tmo[3];

    #pragma unroll
    for (int r = 0; r < 8; ++r) {
        const int n = tn * 16 + 8 * kh + r;
        if (n >= NN) continue;

        const float S1v = accS[r];
        const float Gv  = accG[r];
        const float sp  = sumP[b * NN + n];
        const float f0n = F0[b * NN + n];

        const float cost_dice  = 1.0f - 2.0f * S1v / (sp + c1 + 0.001f);
        const float cost_focal = (f0n + Gv) * (1.0f / (float)KK);
        const float cost_class = -pred_prob[((size_t)b * CC + cls) * NN + n];

        // calc_bbox(pred_momentum)
        const float* pm = pred_momentum + ((size_t)b * NN + n) * 4;
        const float pt  = expf(fminf(pm[0], 10.0f));
        const float rap = TWO_PI_F / (1.0f + expf(-pm[1]));
        const float phi = TWO_PI_F / (1.0f + expf(-pm[2]));
        const float mmv = expf(fminf(pm[3], 10.0f));

        // phase-wrap corrections (component 2 only)
        const float pe2 = (t2 - phi > PI_F) ? TWO_PI_F : 0.0f;  // added to pred
        const float te2 = (phi - t2 > PI_F) ? TWO_PI_F : 0.0f;  // added to tgt

        // momentum L1
        float l1 = fabsf(pt  / (t0 + 0.001f) - 1.0f)
                 + fabsf(rap / (t1 + 0.001f) - 1.0f)
                 + fabsf((phi + pe2) / (t2 + te2 + 0.001f) - 1.0f)
                 + fabsf(mmv / (t3 + 0.001f) - 1.0f);
        l1 *= 0.25f;

        // eIoU: w1=pt, x1=rap, y1=phi, h1=m ; tgt already bbox'd
        const float w1 = pt,  x1 = rap, h1 = mmv;
        const float w2 = t0,  x2 = t1,  h2 = t3;
        const float y1a = phi + pe2;   // y1 + y1e
        const float y2a = t2  + te2;   // y2 + y2e
        const float wI = fminf(x1 + w1 * 0.5f, x2 + w2 * 0.5f)
                       - fmaxf(x1 - w1 * 0.5f, x2 - w2 * 0.5f);
        const float hI = fminf(y1a + h1 * 0.5f, y2a + h2 * 0.5f)
                       - fmaxf(y1a - h1 * 0.5f, y2a - h2 * 0.5f);
        const float iou = ((wI > 0.0f) && (hI > 0.0f))
            ? wI * hI / (w1 * h1 + w2 * h2 - wI * hI + 0.001f) : 0.0f;
        const float cwe = fmaxf(x1 + w1 * 0.5f, x2 + w2 * 0.5f)
                        - fminf(x1 - w1 * 0.5f, x2 - w2 * 0.5f);
        const float che = fmaxf(y1a + h1 * 0.5f, y2a + h2 * 0.5f)
                        - fminf(y1a - h1 * 0.5f, y2a - h2 * 0.5f);
        const float cw2 = cwe * cwe;
        const float ch2 = che * che;
        const float rho_w = (w1 - w2) * (w1 - w2) / (cw2 + 0.001f);
        const float rho_h = (h1 - h2) * (h1 - h2) / (ch2 + 0.001f);
        const float rho_d = ((x1 - x2) * (x1 - x2) + (y1a - y2a) * (y1a - y2a))
                          / (cw2 + ch2 + 0.001f);
        const float ei = 1.0f - iou + rho_d + rho_w + rho_h;

        float cost = 0.25f * cost_focal + 0.75f * cost_dice + cost_class
                   + 0.8f * l1 + 0.2f * ei;
        out[((size_t)b * NN + n) * MM + m] = (cls > 0) ? cost : 100000.0f;
    }
}

// ---------------------------------------------------------------------------
extern "C" void kernel_launch(void* const* d_in, const int* in_sizes, int n_in,
                              void* d_out, int out_size, void* d_ws, size_t ws_size,
                              hipStream_t stream) {
    const float* pred_prob       = (const float*)d_in[0];   // (2,10,100)
    const float* pred_mask       = (const float*)d_in[1];   // (2,100,24,24)
    const float* pred_momentum   = (const float*)d_in[2];   // (2,100,4)
    const int*   target_class    = (const int*)d_in[3];     // (2,50)
    const float* target_mask     = (const float*)d_in[4];   // (2,50,96,96)
    const float* target_momentum = (const float*)d_in[5];   // (2,50,4)
    float* out = (float*)d_out;                             // (2,100,50)

    float* P      = (float*)d_ws;                 // B*N*K
    float* Dm     = P  + (size_t)BB * NN * KK;    // B*N*K
    float* sumP   = Dm + (size_t)BB * NN * KK;    // B*N
    float* F0     = sumP + BB * NN;               // B*N
    float* count1 = F0   + BB * NN;               // B*M

    prep_pred_kernel<<<BB * NN, 256, 0, stream>>>(pred_mask, P, Dm, sumP, F0);
    prep_tgt_kernel<<<BB * MM, 256, 0, stream>>>(target_mask, count1);

    dim3 grid((MM + 15) / 16, (NN + 15) / 16, BB);   // (4,7,2)
    cost_kernel<<<grid, 32, 0, stream>>>(
        pred_prob, pred_momentum, target_class, target_mask, target_momentum,
        P, Dm, sumP, F0, count1, out);
}